// GNN_40759239639518
// MI455X (gfx1250) — compile-verified
//
#include <hip/hip_runtime.h>
#include <math.h>

typedef float v2f __attribute__((ext_vector_type(2)));
typedef float v8f __attribute__((ext_vector_type(8)));

#define F_IN  128
#define H_MID 16
#define C_OUT 10

// ---------------- degree / normalization ----------------

__global__ void k_deg_init(float* deg, int n) {
    int i = blockIdx.x * blockDim.x + threadIdx.x;
    if (i < n) deg[i] = 1.0f;  // self-loop
}

__global__ void k_deg_accum(const int* __restrict__ col, float* deg, int E) {
    int i = blockIdx.x * blockDim.x + threadIdx.x;
    if (i < E) atomicAdd(&deg[col[i]], 1.0f);
}

__global__ void k_deg_to_dinv(float* deg, int n) {
    int i = blockIdx.x * blockDim.x + threadIdx.x;
    if (i < n) deg[i] = rsqrtf(deg[i]);  // deg >= 1 always (self-loops)
}

// ---------------- layer-1 GEMM: h = x @ W1 + b1  (WMMA f32 16x16x4) ----------------

__global__ void k_gemm1_wmma(const float* __restrict__ x, const float* __restrict__ W1,
                             const float* __restrict__ b1, float* __restrict__ h,
                             int n, int ntiles) {
    __shared__ float w[F_IN * H_MID];  // 8 KB of the 320 KB/WGP LDS
    for (int i = threadIdx.x; i < F_IN * H_MID; i += blockDim.x) w[i] = W1[i];
    __syncthreads();

    int wave = threadIdx.x >> 5;
    int lane = threadIdx.x & 31;
    int tile = blockIdx.x * (blockDim.x >> 5) + wave;
    if (tile >= ntiles) return;

    int half = lane >> 4;   // selects K+2 stripe per ISA A/B layout
    int m    = lane & 15;   // A: row index; B/D: column index

    int row = tile * 16 + m;
    int rowc = row < n ? row : n - 1;              // clamp loads for tail tile
    const float* arow = x + (size_t)rowc * F_IN;

    v8f c = {};
#pragma unroll
    for (int kk = 0; kk < F_IN; kk += 4) {
        v2f a, b;
        // A 16x4 f32 layout: VGPR j holds K = kk + j + 2*half for this lane
        a.x = arow[kk + 2 * half + 0];
        a.y = arow[kk + 2 * half + 1];
        // B 4x16 layout mirrors C striping: VGPR j, low lanes K=kk+j, high lanes K=kk+j+2
        b.x = w[(kk + 2 * half + 0) * H_MID + m];
        b.y = w[(kk + 2 * half + 1) * H_MID + m];
        c = __builtin_amdgcn_wmma_f32_16x16x4_f32(false, a, false, b, (short)0, c,
                                                  false, false);
    }

    float bias = b1[m];
#pragma unroll
    for (int r = 0; r < 8; ++r) {
        int orow = tile * 16 + r + 8 * half;       // C/D: VGPR r -> M = r + 8*half
        if (orow < n) h[(size_t)orow * H_MID + m] = c[r] + bias;
    }
}

// ---------------- self-loop term doubles as accumulator init ----------------

__global__ void k_selfloop1(const float* __restrict__ dinv, const float* __restrict__ h,
                            float* __restrict__ agg, int n) {
    int i = blockIdx.x * blockDim.x + threadIdx.x;  // over n*16
    if (i < n * H_MID) {
        int node = i >> 4;
        float d = dinv[node];
        agg[i] = d * d * h[i];
    }
}

__global__ void k_scatter1(const int* __restrict__ srcs, const int* __restrict__ dsts,
                           const float* __restrict__ dinv, const float* __restrict__ h,
                           float* __restrict__ agg, int E) {
    int tid = blockIdx.x * blockDim.x + threadIdx.x;  // over E*16
    if (tid >= E * H_MID) return;
    int e = tid >> 4;
    int f = tid & 15;
    int s = srcs[e];
    int t = dsts[e];
    float norm = dinv[s] * dinv[t];
    atomicAdd(&agg[(size_t)t * H_MID + f], norm * h[(size_t)s * H_MID + f]);
}

// ---------------- layer-2 GEMM: h2 = relu(agg1) @ W2 + b2  (WMMA, N padded to 16) ----

__global__ void k_gemm2_wmma(const float* __restrict__ agg1, const float* __restrict__ W2,
                             const float* __restrict__ b2, float* __restrict__ h2,
                             int n, int ntiles) {
    __shared__ float w[H_MID * 16];  // W2 zero-padded 16x10 -> 16x16
    {
        int k = threadIdx.x >> 4, nn = threadIdx.x & 15;
        if (threadIdx.x < H_MID * 16)
            w[threadIdx.x] = (nn < C_OUT) ? W2[k * C_OUT + nn] : 0.0f;
    }
    __syncthreads();

    int wave = threadIdx.x >> 5;
    int lane = threadIdx.x & 31;
    int tile = blockIdx.x * (blockDim.x >> 5) + wave;
    if (tile >= ntiles) return;

    int half = lane >> 4;
    int m    = lane & 15;

    int row  = tile * 16 + m;
    int rowc = row < n ? row : n - 1;
    const float* arow = agg1 + (size_t)rowc * H_MID;

    v8f c = {};
#pragma unroll
    for (int kk = 0; kk < H_MID; kk += 4) {
        v2f a, b;
        a.x = fmaxf(arow[kk + 2 * half + 0], 0.0f);  // fused ReLU
        a.y = fmaxf(arow[kk + 2 * half + 1], 0.0f);
        b.x = w[(kk + 2 * half + 0) * 16 + m];
        b.y = w[(kk + 2 * half + 1) * 16 + m];
        c = __builtin_amdgcn_wmma_f32_16x16x4_f32(false, a, false, b, (short)0, c,
                                                  false, false);
    }

    if (m < C_OUT) {
        float bias = b2[m];
#pragma unroll
        for (int r = 0; r < 8; ++r) {
            int orow = tile * 16 + r + 8 * half;
            if (orow < n) h2[(size_t)orow * C_OUT + m] = c[r] + bias;
        }
    }
}

__global__ void k_selfloop2(const float* __restrict__ dinv, const float* __restrict__ h2,
                            float* __restrict__ agg2, int n) {
    int i = blockIdx.x * blockDim.x + threadIdx.x;  // over nodes
    if (i < n) {
        float d2 = dinv[i] * dinv[i];
#pragma unroll
        for (int j = 0; j < C_OUT; ++j)
            agg2[(size_t)i * C_OUT + j] = d2 * h2[(size_t)i * C_OUT + j];
    }
}

__global__ void k_scatter2(const int* __restrict__ srcs, const int* __restrict__ dsts,
                           const float* __restrict__ dinv, const float* __restrict__ h2,
                           float* __restrict__ agg2, int E) {
    int tid = blockIdx.x * blockDim.x + threadIdx.x;  // over E*16, feats 10..15 idle
    if (tid >= E * 16) return;
    int e = tid >> 4;
    int f = tid & 15;
    if (f >= C_OUT) return;
    int s = srcs[e];
    int t = dsts[e];
    float norm = dinv[s] * dinv[t];
    atomicAdd(&agg2[(size_t)t * C_OUT + f], norm * h2[(size_t)s * C_OUT + f]);
}

__global__ void k_log_softmax(const float* __restrict__ agg2, float* __restrict__ out,
                              int n) {
    int i = blockIdx.x * blockDim.x + threadIdx.x;
    if (i >= n) return;
    float v[C_OUT];
    float mx = -INFINITY;
#pragma unroll
    for (int j = 0; j < C_OUT; ++j) {
        v[j] = agg2[(size_t)i * C_OUT + j];
        mx = fmaxf(mx, v[j]);
    }
    float s = 0.0f;
#pragma unroll
    for (int j = 0; j < C_OUT; ++j) s += expf(v[j] - mx);
    float ls = logf(s);
#pragma unroll
    for (int j = 0; j < C_OUT; ++j)
        out[(size_t)i * C_OUT + j] = v[j] - mx - ls;
}

// ---------------- launcher ----------------

extern "C" void kernel_launch(void* const* d_in, const int* in_sizes, int n_in,
                              void* d_out, int out_size, void* d_ws, size_t ws_size,
                              hipStream_t stream) {
    (void)n_in; (void)out_size; (void)ws_size;
    const float* x  = (const float*)d_in[0];
    const int*   ei = (const int*)d_in[1];   // [2, E] flat: rows then cols
    const float* W1 = (const float*)d_in[2];
    const float* b1 = (const float*)d_in[3];
    const float* W2 = (const float*)d_in[4];
    const float* b2 = (const float*)d_in[5];
    float* out = (float*)d_out;

    const int n = in_sizes[0] / F_IN;       // 100000
    const int E = in_sizes[1] / 2;          // 3200000
    const int* srcs = ei;
    const int* dsts = ei + E;

    // workspace layout (floats): dinv[n] | h1[n*16] (aliased by h2[n*10]) | agg1[n*16] | agg2[n*10]
    float* ws   = (float*)d_ws;
    float* dinv = ws;
    float* h1   = dinv + n;
    float* agg1 = h1 + (size_t)n * H_MID;
    float* h2   = h1;                        // h1 dead once gemm2 reads agg1
    float* agg2 = agg1 + (size_t)n * H_MID;

    const int B = 256;
    int ntiles = (n + 15) / 16;
    int gemm_blocks = (ntiles + (B / 32) - 1) / (B / 32);

    k_deg_init<<<(n + B - 1) / B, B, 0, stream>>>(dinv, n);
    k_deg_accum<<<(E + B - 1) / B, B, 0, stream>>>(dsts, dinv, E);
    k_deg_to_dinv<<<(n + B - 1) / B, B, 0, stream>>>(dinv, n);

    k_gemm1_wmma<<<gemm_blocks, B, 0, stream>>>(x, W1, b1, h1, n, ntiles);
    k_selfloop1<<<((size_t)n * H_MID + B - 1) / B, B, 0, stream>>>(dinv, h1, agg1, n);
    k_scatter1<<<((size_t)E * H_MID + B - 1) / B, B, 0, stream>>>(srcs, dsts, dinv, h1, agg1, E);

    k_gemm2_wmma<<<gemm_blocks, B, 0, stream>>>(agg1, W2, b2, h2, n, ntiles);
    k_selfloop2<<<(n + B - 1) / B, B, 0, stream>>>(dinv, h2, agg2, n);
    k_scatter2<<<((size_t)E * 16 + B - 1) / B, B, 0, stream>>>(srcs, dsts, dinv, h2, agg2, E);

    k_log_softmax<<<(n + B - 1) / B, B, 0, stream>>>(agg2, out, n);
}